// Model3_31233002176865
// MI455X (gfx1250) — compile-verified
//
#include <hip/hip_runtime.h>
#include <cstdint>

typedef __attribute__((ext_vector_type(2))) float v2f;
typedef __attribute__((ext_vector_type(8))) float v8f;

#define CODE 2048
#define ARR  512
#define LENS 16384
#define NANG 181
#define NPAD 192
#define KCONST 0.1f

// Workspace layout (floats):
//   [0, LENS)                    theta_norm_re
//   [LENS, 2*LENS)               theta_norm_im
//   [2*LENS, 2*LENS+ARR*NPAD)    M_re   (zero-padded cols 181..191)
//   [.. +ARR*NPAD)               M_im
#define WS_TNRE 0
#define WS_TNIM (LENS)
#define WS_MRE  (2*LENS)
#define WS_MIM  (2*LENS + ARR*NPAD)

#define WMMA_F32(A, B, C) \
  __builtin_amdgcn_wmma_f32_16x16x4_f32(false, (A), false, (B), (short)0, (C), false, false)

// ---------------------------------------------------------------------------
// Init: normalize theta, zero M accumulator and output scalar (deterministic
// per call; harness poisons ws/out and never re-poisons between replays).
// ---------------------------------------------------------------------------
__global__ void __launch_bounds__(256)
k_init(const float* __restrict__ theta_re, const float* __restrict__ theta_im,
       float* __restrict__ ws, float* __restrict__ out) {
  int tid = blockIdx.x * blockDim.x + threadIdx.x;
  if (tid < LENS) {
    float tr = theta_re[tid], ti = theta_im[tid];
    float inv = rsqrtf(tr * tr + ti * ti);
    ws[WS_TNRE + tid] = tr * inv;
    ws[WS_TNIM + tid] = ti * inv;
  }
  if (tid < ARR * NPAD) {
    ws[WS_MRE + tid] = 0.0f;
    ws[WS_MIM + tid] = 0.0f;
  }
  if (tid == 0) out[0] = 0.0f;
}

// ---------------------------------------------------------------------------
// Kernel A: M[a,n] = sum_l (G[a,l] * theta_n[l]) * steer[l,n]   (complex)
// Per-wave job: TWO 16x16 output tiles (shared A fragments) over a 2048-long
// K slice. 32 mTiles x 6 nPairs x 8 K-splits = 1536 wave-jobs.
// Complex product via 3-WMMA Karatsuba: P1=ArBr, P2=AiBi, P3=(Ar+Ai)(Br+Bi).
// ---------------------------------------------------------------------------
__global__ void __launch_bounds__(256)
k_lens(const float* __restrict__ G_re, const float* __restrict__ G_im,
       const float* __restrict__ steer_re, const float* __restrict__ steer_im,
       float* __restrict__ ws) {
  const float* tn_re = ws + WS_TNRE;
  const float* tn_im = ws + WS_TNIM;
  float* M_re = ws + WS_MRE;
  float* M_im = ws + WS_MIM;

  const int wave = (blockIdx.x * blockDim.x + threadIdx.x) >> 5;
  const int lane = threadIdx.x & 31;
  const int kS   = wave & 7;
  const int rest = wave >> 3;
  const int nP   = rest % 6;       // n-pair: covers columns [nP*32, nP*32+32)
  const int mT   = rest / 6;

  const int m    = lane & 15;      // A row within tile
  const int half = lane >> 4;      // K sub-pair selector
  const int n0   = (nP << 5) + (lane & 15);        // always < 176 < NANG
  const int n1   = n0 + 16;                        // up to 191: may be invalid
  const float mask1 = (n1 < NANG) ? 1.0f : 0.0f;   // branchless tail guard
  const int n1c  = (n1 < NANG) ? n1 : 0;           // clamped (safe) address

  const int l0 = kS * (LENS / 8);
  const int kSteps = (LENS / 8) / 4;

  const float* pGr = G_re + (size_t)((mT << 4) + m) * LENS + l0 + 2 * half;
  const float* pGi = G_im + (size_t)((mT << 4) + m) * LENS + l0 + 2 * half;
  const float* pTr = tn_re + l0 + 2 * half;
  const float* pTi = tn_im + l0 + 2 * half;
  const float* pSr0 = steer_re + (size_t)(l0 + 2 * half) * NANG + n0;
  const float* pSi0 = steer_im + (size_t)(l0 + 2 * half) * NANG + n0;
  const float* pSr1 = steer_re + (size_t)(l0 + 2 * half) * NANG + n1c;
  const float* pSi1 = steer_im + (size_t)(l0 + 2 * half) * NANG + n1c;

  v8f p1_0 = {}, p2_0 = {}, p3_0 = {};
  v8f p1_1 = {}, p2_1 = {}, p3_1 = {};

  for (int it = 0; it < kSteps; ++it) {
    v2f gr = *(const v2f*)pGr;
    v2f gi = *(const v2f*)pGi;
    v2f tr = *(const v2f*)pTr;
    v2f ti = *(const v2f*)pTi;
    // A = G * theta_norm (complex, elementwise over K columns); shared by both tiles
    v2f aRe = gr * tr - gi * ti;
    v2f aIm = gr * ti + gi * tr;
    v2f aS  = aRe + aIm;

    // tile 0 (always valid)
    v2f bRe0, bIm0;
    bRe0.x = pSr0[0]; bRe0.y = pSr0[NANG];
    bIm0.x = pSi0[0]; bIm0.y = pSi0[NANG];
    v2f bS0 = bRe0 + bIm0;
    p1_0 = WMMA_F32(aRe, bRe0, p1_0);
    p2_0 = WMMA_F32(aIm, bIm0, p2_0);
    p3_0 = WMMA_F32(aS,  bS0,  p3_0);

    // tile 1 (tail columns masked to zero; address clamped, load unconditional)
    v2f bRe1, bIm1;
    bRe1.x = pSr1[0]; bRe1.y = pSr1[NANG];
    bIm1.x = pSi1[0]; bIm1.y = pSi1[NANG];
    bRe1 *= mask1; bIm1 *= mask1;
    v2f bS1 = bRe1 + bIm1;
    p1_1 = WMMA_F32(aRe, bRe1, p1_1);
    p2_1 = WMMA_F32(aIm, bIm1, p2_1);
    p3_1 = WMMA_F32(aS,  bS1,  p3_1);

    pGr += 4; pGi += 4; pTr += 4; pTi += 4;
    pSr0 += 4 * NANG; pSi0 += 4 * NANG;
    pSr1 += 4 * NANG; pSi1 += 4 * NANG;
  }

  // Karatsuba recombination: Re = P1 - P2, Im = P3 - P1 - P2
  v8f re0 = p1_0 - p2_0;
  v8f im0 = p3_0 - p1_0 - p2_0;
  v8f re1 = p1_1 - p2_1;
  v8f im1 = p3_1 - p1_1 - p2_1;

  // C/D layout: lane&15 = N col, VGPR v -> M row = v + 8*half
  const int col0 = (nP << 5) + (lane & 15);
#pragma unroll
  for (int v = 0; v < 8; ++v) {
    const int row = (mT << 4) + v + 8 * half;
    const int idx0 = row * NPAD + col0;
    unsafeAtomicAdd(&M_re[idx0], re0[v]);
    unsafeAtomicAdd(&M_im[idx0], im0[v]);
    unsafeAtomicAdd(&M_re[idx0 + 16], re1[v]);   // masked tiles add 0.0
    unsafeAtomicAdd(&M_im[idx0 + 16], im1[v]);
  }
}

// ---------------------------------------------------------------------------
// Kernel B: sout = (w/|w|) @ M, fused reduction:
//   out = K * sum|sout|^2 - (1+2K) * sum_c |sout[c, idx_c]|^2
// 128 cTiles x 12 nTiles = 1536 wave-jobs.
// ---------------------------------------------------------------------------
__global__ void __launch_bounds__(256)
k_sout(const float* __restrict__ w_re, const float* __restrict__ w_im,
       const int* __restrict__ sweep_angle,
       const float* __restrict__ ws, float* __restrict__ out) {
  const float* M_re = ws + WS_MRE;
  const float* M_im = ws + WS_MIM;

  const int wave = (blockIdx.x * blockDim.x + threadIdx.x) >> 5;
  const int lane = threadIdx.x & 31;
  const int nT = wave % 12;
  const int cT = wave / 12;

  const int m    = lane & 15;
  const int half = lane >> 4;
  const int n    = (nT << 4) + (lane & 15);   // 0..191; M is zero-padded

  const float* pWr = w_re + (size_t)((cT << 4) + m) * ARR + 2 * half;
  const float* pWi = w_im + (size_t)((cT << 4) + m) * ARR + 2 * half;
  const float* pMr = M_re + (size_t)(2 * half) * NPAD + n;
  const float* pMi = M_im + (size_t)(2 * half) * NPAD + n;

  v8f p1 = {}, p2 = {}, p3 = {};

  for (int k = 0; k < ARR; k += 4) {
    v2f wr = *(const v2f*)pWr;
    v2f wi = *(const v2f*)pWi;
    v2f inv;
    inv.x = rsqrtf(wr.x * wr.x + wi.x * wi.x);
    inv.y = rsqrtf(wr.y * wr.y + wi.y * wi.y);
    v2f aRe = wr * inv;               // unit-modulus weights
    v2f aIm = wi * inv;
    v2f aS  = aRe + aIm;
    v2f bRe, bIm;
    bRe.x = pMr[0]; bRe.y = pMr[NPAD];
    bIm.x = pMi[0]; bIm.y = pMi[NPAD];
    v2f bS = bRe + bIm;
    p1 = WMMA_F32(aRe, bRe, p1);
    p2 = WMMA_F32(aIm, bIm, p2);
    p3 = WMMA_F32(aS,  bS,  p3);
    pWr += 4; pWi += 4;
    pMr += 4 * NPAD; pMi += 4 * NPAD;
  }

  v8f sRe = p1 - p2;
  v8f sIm = p3 - p1 - p2;

  // Fused epilogue: per-element coefficient K, diagonal gets K-(1+2K) = -(1+K)
  float partial = 0.0f;
  const bool nValid = (n < NANG);
#pragma unroll
  for (int v = 0; v < 8; ++v) {
    const int row = (cT << 4) + v + 8 * half;
    const float p = sRe[v] * sRe[v] + sIm[v] * sIm[v];
    const int idx = sweep_angle[row] + 90;
    const float coef = nValid ? (KCONST + ((n == idx) ? -(1.0f + 2.0f * KCONST) : 0.0f)) : 0.0f;
    partial += coef * p;
  }
  // wave32 butterfly reduction, one atomic per wave
  for (int off = 16; off >= 1; off >>= 1)
    partial += __shfl_xor(partial, off, 32);
  if (lane == 0) unsafeAtomicAdd(out, partial);
}

// ---------------------------------------------------------------------------
extern "C" void kernel_launch(void* const* d_in, const int* in_sizes, int n_in,
                              void* d_out, int out_size, void* d_ws, size_t ws_size,
                              hipStream_t stream) {
  const float* w_re     = (const float*)d_in[0];
  const float* w_im     = (const float*)d_in[1];
  const float* theta_re = (const float*)d_in[2];
  const float* theta_im = (const float*)d_in[3];
  const float* G_re     = (const float*)d_in[4];
  const float* G_im     = (const float*)d_in[5];
  const float* steer_re = (const float*)d_in[6];
  const float* steer_im = (const float*)d_in[7];
  // addw_re/addw_im (d_in[8], d_in[9]) only feed the 0-weighted variance term.
  const int*   sweep    = (const int*)d_in[10];

  float* ws  = (float*)d_ws;
  float* out = (float*)d_out;

  // init: cover max(LENS, ARR*NPAD) = 98304 elements
  k_init<<<(ARR * NPAD + 255) / 256, 256, 0, stream>>>(theta_re, theta_im, ws, out);

  // kernel A: 1536 wave-jobs (2 n-tiles each), 8 waves/block
  k_lens<<<1536 / 8, 256, 0, stream>>>(G_re, G_im, steer_re, steer_im, ws);

  // kernel B: 1536 wave-jobs, 8 waves/block
  k_sout<<<1536 / 8, 256, 0, stream>>>(w_re, w_im, sweep, ws, out);
}